// VanillaRNN_47588237639944
// MI455X (gfx1250) — compile-verified
//
#include <hip/hip_runtime.h>
#include <hip/hip_bf16.h>
#include <math.h>

// Shapes (fixed by the reference)
#define VOCAB   128
#define EMBED   256
#define HIDDEN  512
#define BATCH   64
#define SEQ     512
#define BT      (BATCH * SEQ)          // 32768 rows
#define LROW    516                    // padded LDS row stride (bank-conflict free)

typedef __attribute__((ext_vector_type(2))) float v2f;
typedef __attribute__((ext_vector_type(8))) float v8f;

// fp32 WMMA: D(16x16) = A(16x4) * B(4x16) + C
__device__ __forceinline__ v8f wmma4(v2f a, v2f b, v8f c) {
  return __builtin_amdgcn_wmma_f32_16x16x4_f32(
      /*neg_a=*/false, a, /*neg_b=*/false, b,
      /*c_mod=*/(short)0, c, /*reuse_a=*/false, /*reuse_b=*/false);
}

// K-loop for two M-tiles sharing the B fragments: 2 A-loads + NTile B-loads
// feed 2*NTile WMMAs per k-step.
template<int K, int NTile>
__device__ __forceinline__ void mma_rows2(v8f acc0[NTile], v8f acc1[NTile],
                                          const float* __restrict__ arow0,
                                          const float* __restrict__ arow1,
                                          const float* const wrow[NTile],
                                          int half) {
  for (int k = 0; k < K; k += 4) {
    v2f a0 = *(const v2f*)(arow0 + k + half * 2);
    v2f a1 = *(const v2f*)(arow1 + k + half * 2);
#pragma unroll
    for (int j = 0; j < NTile; ++j) {
      v2f b = *(const v2f*)(wrow[j] + k + half * 2);
      acc0[j] = wmma4(a0, b, acc0[j]);
      acc1[j] = wmma4(a1, b, acc1[j]);
    }
  }
}

// ---------------------------------------------------------------------------
// Layer-0 input projection, fused embedding gather:
//   out[m][n] = sum_k emb[x[m]][k] * Wxh[n][k] + bias[n]
//   M = BT, K = EMBED, N = HIDDEN.  MT=2 m-tiles, NT=4 n-tiles per wave.
// ---------------------------------------------------------------------------
__global__ __launch_bounds__(256) void gather_proj_kernel(
    const int* __restrict__ x, const float* __restrict__ emb,
    const float* __restrict__ Wxh, const float* __restrict__ bias,
    float* __restrict__ out)
{
  constexpr int NTile = 4;
  const int ngroups = HIDDEN / (16 * NTile);              // 8
  const int wid   = blockIdx.x * (blockDim.x >> 5) + (threadIdx.x >> 5);
  const int mpair = wid / ngroups;
  const int grp   = wid % ngroups;
  const int lane  = threadIdx.x & 31;
  const int half  = lane >> 4;
  const int lm    = lane & 15;

  const int m0 = mpair * 32 + lm;
  const int m1 = m0 + 16;
  const float* arow0 = emb + (size_t)x[m0] * EMBED;
  const float* arow1 = emb + (size_t)x[m1] * EMBED;
  const float* wrow[NTile];
#pragma unroll
  for (int j = 0; j < NTile; ++j)
    wrow[j] = Wxh + (size_t)((grp * NTile + j) * 16 + lm) * EMBED;

  __builtin_prefetch(arow0, 0, 0);
  __builtin_prefetch(wrow[0], 0, 0);

  v8f acc0[NTile] = {}, acc1[NTile] = {};
  mma_rows2<EMBED, NTile>(acc0, acc1, arow0, arow1, wrow, half);

#pragma unroll
  for (int j = 0; j < NTile; ++j) {
    const int n  = (grp * NTile + j) * 16 + lm;
    const float bv = bias[n];
#pragma unroll
    for (int v = 0; v < 8; ++v) {
      const int r = half * 8 + v;
      out[(size_t)(mpair * 32 + r)      * HIDDEN + n] = acc0[j][v] + bv;
      out[(size_t)(mpair * 32 + 16 + r) * HIDDEN + n] = acc1[j][v] + bv;
    }
  }
}

// ---------------------------------------------------------------------------
// Plain projection:  out[m][n] = sum_k in[m][k] * W[n][k] + bias[n]
//   M = BT, K = HIDDEN, N = HIDDEN
// ---------------------------------------------------------------------------
__global__ __launch_bounds__(256) void proj_kernel(
    const float* __restrict__ in, const float* __restrict__ W,
    const float* __restrict__ bias, float* __restrict__ out)
{
  constexpr int NTile = 4;
  const int ngroups = HIDDEN / (16 * NTile);              // 8
  const int wid   = blockIdx.x * (blockDim.x >> 5) + (threadIdx.x >> 5);
  const int mpair = wid / ngroups;
  const int grp   = wid % ngroups;
  const int lane  = threadIdx.x & 31;
  const int half  = lane >> 4;
  const int lm    = lane & 15;

  const float* arow0 = in + (size_t)(mpair * 32 + lm)      * HIDDEN;
  const float* arow1 = in + (size_t)(mpair * 32 + 16 + lm) * HIDDEN;
  const float* wrow[NTile];
#pragma unroll
  for (int j = 0; j < NTile; ++j)
    wrow[j] = W + (size_t)((grp * NTile + j) * 16 + lm) * HIDDEN;

  __builtin_prefetch(arow0, 0, 0);
  __builtin_prefetch(wrow[0], 0, 0);

  v8f acc0[NTile] = {}, acc1[NTile] = {};
  mma_rows2<HIDDEN, NTile>(acc0, acc1, arow0, arow1, wrow, half);

#pragma unroll
  for (int j = 0; j < NTile; ++j) {
    const int n  = (grp * NTile + j) * 16 + lm;
    const float bv = bias[n];
#pragma unroll
    for (int v = 0; v < 8; ++v) {
      const int r = half * 8 + v;
      out[(size_t)(mpair * 32 + r)      * HIDDEN + n] = acc0[j][v] + bv;
      out[(size_t)(mpair * 32 + 16 + r) * HIDDEN + n] = acc1[j][v] + bv;
    }
  }
}

// ---------------------------------------------------------------------------
// Persistent recurrence for one layer.  Each block owns one 16-row batch
// tile for the whole scan; h stays resident in LDS (row stride 516 floats ->
// conflict-free ds_load_b64 of A fragments).  One block = 16 waves, each wave
// covers 2 n-tiles (32 of the 512 hidden columns).
//   h_t[b][n] = tanh( xW[b][t][n] + sum_k h_{t-1}[b][k] * Whh[n][k] )
// ---------------------------------------------------------------------------
__global__ __launch_bounds__(512) void rnn_layer_kernel(
    const float* __restrict__ xW, const float* __restrict__ Whh,
    float* __restrict__ hs)
{
  constexpr int NTile = 2;
  __shared__ float hbuf[16 * LROW];                       // 33 KB

  const int warp = threadIdx.x >> 5;                      // 0..15 n-group
  const int lane = threadIdx.x & 31;
  const int half = lane >> 4;
  const int lm   = lane & 15;
  const int tile_m = blockIdx.x;                          // 0..3

  const float* wrow[NTile];
#pragma unroll
  for (int j = 0; j < NTile; ++j)
    wrow[j] = Whh + (size_t)((warp * NTile + j) * 16 + lm) * HIDDEN;

  __builtin_prefetch(wrow[0], 0, 0);

  // h_{-1} = 0
  for (int i = threadIdx.x; i < 16 * LROW; i += blockDim.x) hbuf[i] = 0.f;
  __syncthreads();

  const float* arow = hbuf + lm * LROW;                   // lane's batch row

  for (int t = 0; t < SEQ; ++t) {
    v8f acc[NTile] = {};
    for (int k = 0; k < HIDDEN; k += 4) {
      v2f a = *(const v2f*)(arow + k + half * 2);         // ds_load_b64
#pragma unroll
      for (int j = 0; j < NTile; ++j) {
        v2f b = *(const v2f*)(wrow[j] + k + half * 2);
        acc[j] = wmma4(a, b, acc[j]);
      }
    }
    __syncthreads();                                      // h_{t-1} reads done

#pragma unroll
    for (int j = 0; j < NTile; ++j) {
      const int n = (warp * NTile + j) * 16 + lm;
#pragma unroll
      for (int v = 0; v < 8; ++v) {
        const int r  = half * 8 + v;                      // local batch row
        const int mm = tile_m * 16 + r;
        const size_t o = (size_t)mm * (SEQ * HIDDEN) + (size_t)t * HIDDEN + n;
        const float hv = tanhf(acc[j][v] + xW[o]);
        hbuf[r * LROW + n] = hv;                          // next step's A
        hs[o] = hv;                                       // layer output
      }
    }
    __syncthreads();                                      // h_t visible
  }
}

// ---------------------------------------------------------------------------
// Output projection: logits[m][n] = sum_k hs1[m][k] * Why[n][k] + by[n]
//   M = BT, K = HIDDEN, N = VOCAB
// ---------------------------------------------------------------------------
__global__ __launch_bounds__(256) void logits_kernel(
    const float* __restrict__ hs, const float* __restrict__ Why,
    const float* __restrict__ by, float* __restrict__ out)
{
  constexpr int NTile = 4;
  const int ngroups = VOCAB / (16 * NTile);               // 2
  const int wid   = blockIdx.x * (blockDim.x >> 5) + (threadIdx.x >> 5);
  const int mpair = wid / ngroups;
  const int grp   = wid % ngroups;
  const int lane  = threadIdx.x & 31;
  const int half  = lane >> 4;
  const int lm    = lane & 15;

  const float* arow0 = hs + (size_t)(mpair * 32 + lm)      * HIDDEN;
  const float* arow1 = hs + (size_t)(mpair * 32 + 16 + lm) * HIDDEN;
  const float* wrow[NTile];
#pragma unroll
  for (int j = 0; j < NTile; ++j)
    wrow[j] = Why + (size_t)((grp * NTile + j) * 16 + lm) * HIDDEN;

  __builtin_prefetch(arow0, 0, 0);
  __builtin_prefetch(wrow[0], 0, 0);

  v8f acc0[NTile] = {}, acc1[NTile] = {};
  mma_rows2<HIDDEN, NTile>(acc0, acc1, arow0, arow1, wrow, half);

#pragma unroll
  for (int j = 0; j < NTile; ++j) {
    const int n  = (grp * NTile + j) * 16 + lm;
    const float bv = by[n];
#pragma unroll
    for (int v = 0; v < 8; ++v) {
      const int r = half * 8 + v;
      out[(size_t)(mpair * 32 + r)      * VOCAB + n] = acc0[j][v] + bv;
      out[(size_t)(mpair * 32 + 16 + r) * VOCAB + n] = acc1[j][v] + bv;
    }
  }
}

// Extract h_last = hs[:, T-1, :]  -> dst (B, H)
__global__ __launch_bounds__(256) void copy_last_kernel(
    const float* __restrict__ hs, float* __restrict__ dst)
{
  const int i = blockIdx.x * blockDim.x + threadIdx.x;
  if (i < BATCH * HIDDEN) {
    const int b = i / HIDDEN, h = i % HIDDEN;
    dst[i] = hs[(size_t)b * SEQ * HIDDEN + (size_t)(SEQ - 1) * HIDDEN + h];
  }
}

// ---------------------------------------------------------------------------
extern "C" void kernel_launch(void* const* d_in, const int* in_sizes, int n_in,
                              void* d_out, int out_size, void* d_ws, size_t ws_size,
                              hipStream_t stream) {
  (void)in_sizes; (void)n_in; (void)out_size; (void)ws_size;

  const int*   x    = (const int*)  d_in[0];
  const float* emb  = (const float*)d_in[1];
  const float* Wxh0 = (const float*)d_in[2];
  const float* Whh0 = (const float*)d_in[3];
  const float* bh0  = (const float*)d_in[4];
  const float* Wxh1 = (const float*)d_in[5];
  const float* Whh1 = (const float*)d_in[6];
  const float* bh1  = (const float*)d_in[7];
  const float* Why  = (const float*)d_in[8];
  const float* by   = (const float*)d_in[9];
  float* out = (float*)d_out;

  float* xW = (float*)d_ws;                              // (BT, H)  64 MB
  float* hs = xW + (size_t)BT * HIDDEN;                  // (B, T, H) 64 MB

  const int proj_blocks  = ((BT / 32) * (HIDDEN / 64)) / 8;   // 1024
  const int logit_blocks = ((BT / 32) * (VOCAB  / 64)) / 8;   // 256

  // Layer 0: fused gather + input projection
  gather_proj_kernel<<<proj_blocks, 256, 0, stream>>>(x, emb, Wxh0, bh0, xW);
  // Layer 0 recurrence: one persistent kernel, whole scan in LDS
  rnn_layer_kernel<<<BATCH / 16, 512, 0, stream>>>(xW, Whh0, hs);
  // hidden[0] = hs0[:, T-1, :]
  float* hid0 = out + (size_t)BT * VOCAB;
  copy_last_kernel<<<(BATCH * HIDDEN + 255) / 256, 256, 0, stream>>>(hs, hid0);

  // Layer 1 input projection (xW buffer reused; hs0 consumed here)
  proj_kernel<<<proj_blocks, 256, 0, stream>>>(hs, Wxh1, bh1, xW);
  // Layer 1 recurrence, in place over hs (hs0 values are dead)
  rnn_layer_kernel<<<BATCH / 16, 512, 0, stream>>>(xW, Whh1, hs);
  // hidden[1]
  copy_last_kernel<<<(BATCH * HIDDEN + 255) / 256, 256, 0, stream>>>(hs, hid0 + BATCH * HIDDEN);

  // Output projection
  logits_kernel<<<logit_blocks, 256, 0, stream>>>(hs, Why, by, out);
}